// PolyConv_26542897889791
// MI455X (gfx1250) — compile-verified
//
#include <hip/hip_runtime.h>

#define D_FEAT 64
#define TILE_E 512      // edges staged per LDS tile (multiple of block size)
#define SCAT_BLOCK 256  // 8 wave32 waves

// ---------------------------------------------------------------------------
// Kernel 1: in-degree via f32 atomics (exact for deg << 2^24)
// ---------------------------------------------------------------------------
__global__ __launch_bounds__(256) void pc_deg(const int* __restrict__ dst,
                                              float* __restrict__ deg,
                                              int n_edges) {
  int i = blockIdx.x * blockDim.x + threadIdx.x;
  if (i < n_edges) atomicAdd(&deg[dst[i]], 1.0f);
}

// ---------------------------------------------------------------------------
// Kernel 2: deg -> clip(deg,1)^-0.5, in place
// ---------------------------------------------------------------------------
__global__ __launch_bounds__(256) void pc_dinv(float* __restrict__ deg,
                                               int n_nodes) {
  int i = blockIdx.x * blockDim.x + threadIdx.x;
  if (i < n_nodes) {
    float d = fmaxf(deg[i], 1.0f);
    deg[i] = 1.0f / sqrtf(d);
  }
}

// ---------------------------------------------------------------------------
// Kernel 3: edge scatter with CDNA5 async index staging.
//   agg[dst*64 + f..f+3] += feat[src*64 + f..f+3] * dinv[src]
// Each block walks tiles of TILE_E edges (grid-stride).  The src/dst index
// streams for the NEXT tile are DMA'd into LDS with
// global_load_async_to_lds_b32 (ASYNCcnt) while the CURRENT tile's
// gather + global_atomic_add_f32 work executes.  Index reads during
// processing are LDS broadcasts instead of 16x-redundant VMEM loads, and a
// global_prefetch pulls the next chunk's gathered feature row.
// ---------------------------------------------------------------------------
__global__ __launch_bounds__(SCAT_BLOCK) void pc_scatter(
    const float* __restrict__ feat, const float* __restrict__ dinv,
    const int* __restrict__ src, const int* __restrict__ dst,
    float* __restrict__ agg, int n_edges) {
  __shared__ int lds_idx[2][2][TILE_E];  // [buf][0=src,1=dst][edge]

  const int ntiles = (n_edges + TILE_E - 1) / TILE_E;
  int tile = blockIdx.x;
  const int tstride = gridDim.x;
  if (tile >= ntiles) return;  // uniform per block: no barrier hazard

  // Stage one tile's indices into LDS buffer b via async global->LDS copies.
  // Loop trip count is uniform (TILE_E % SCAT_BLOCK == 0) so EXEC stays full
  // and per-wave ASYNCcnt accounting is deterministic; tail edges clamped.
  auto stage = [&](int t, int b) {
    const int tb = t * TILE_E;
    for (int i = threadIdx.x; i < TILE_E; i += SCAT_BLOCK) {
      int ge = tb + i;
      ge = (ge < n_edges) ? ge : (n_edges - 1);
      const int* gs = src + ge;
      const int* gd = dst + ge;
      // low 32 bits of a generic pointer to LDS == wave-relative LDS offset
      unsigned ls = (unsigned)(size_t)&lds_idx[b][0][i];
      unsigned ld = (unsigned)(size_t)&lds_idx[b][1][i];
      asm volatile("global_load_async_to_lds_b32 %0, %1, off"
                   :: "v"(ls), "v"(gs) : "memory");
      asm volatile("global_load_async_to_lds_b32 %0, %1, off"
                   :: "v"(ld), "v"(gd) : "memory");
    }
  };

  stage(tile, 0);
  int b = 0;
  while (tile < ntiles) {
    // my async stages for buffer b have landed in LDS
    asm volatile("s_wait_asynccnt 0x0" ::: "memory");
    // all waves' stages visible; all waves done reading buffer b^1
    __syncthreads();

    const int tn = tile + tstride;
    if (tn < ntiles) stage(tn, b ^ 1);  // overlaps with processing below

    const int tb = tile * TILE_E;
    const int nvalid = (n_edges - tb < TILE_E) ? (n_edges - tb) : TILE_E;
    const int nchunk = nvalid << 4;  // 16 float4-chunks per edge
    for (int c = threadIdx.x; c < nchunk; c += SCAT_BLOCK) {
      const int el = c >> 4;
      const int f = (c & 15) << 2;
      const int s = lds_idx[b][0][el];
      const int d = lds_idx[b][1][el];
      const int cn = c + SCAT_BLOCK;
      if (cn < nchunk) {  // prefetch next chunk's gathered row (gfx1250 path)
        const int s2 = lds_idx[b][0][cn >> 4];
        __builtin_prefetch(feat + (size_t)s2 * D_FEAT + ((cn & 15) << 2), 0, 3);
      }
      float4 v =
          *reinterpret_cast<const float4*>(feat + (size_t)s * D_FEAT + f);
      const float sc = dinv[s];
      float* o = agg + (size_t)d * D_FEAT + f;
      atomicAdd(o + 0, v.x * sc);
      atomicAdd(o + 1, v.y * sc);
      atomicAdd(o + 2, v.z * sc);
      atomicAdd(o + 3, v.w * sc);
    }
    tile = tn;
    b ^= 1;
  }
}

// ---------------------------------------------------------------------------
// Kernel 4: fused streaming update (pure bandwidth pass, float4):
//   fnew = fin - agg * dinv[node]; feat_out = fnew (in-place safe)
//   k==1:  h = theta0*fin + theta1*fnew   (fully initializes d_out)
//   k>1 :  h += thetak*fnew
// ---------------------------------------------------------------------------
__global__ __launch_bounds__(256) void pc_update(
    const float* __restrict__ feat_in, const float* __restrict__ agg,
    const float* __restrict__ dinv, float* __restrict__ feat_out,
    float* __restrict__ h, float theta_prev, float theta_k, int first,
    int n_items) {
  int idx = blockIdx.x * blockDim.x + threadIdx.x;
  if (idx >= n_items) return;
  float di = dinv[idx >> 4];  // 16 lanes share a node -> broadcast load
  float4 fi = reinterpret_cast<const float4*>(feat_in)[idx];
  float4 a = reinterpret_cast<const float4*>(agg)[idx];
  float4 fn;
  fn.x = fi.x - a.x * di;
  fn.y = fi.y - a.y * di;
  fn.z = fi.z - a.z * di;
  fn.w = fi.w - a.w * di;
  reinterpret_cast<float4*>(feat_out)[idx] = fn;
  float4* hp = reinterpret_cast<float4*>(h) + idx;
  if (first) {
    float4 hv;
    hv.x = theta_prev * fi.x + theta_k * fn.x;
    hv.y = theta_prev * fi.y + theta_k * fn.y;
    hv.z = theta_prev * fi.z + theta_k * fn.z;
    hv.w = theta_prev * fi.w + theta_k * fn.w;
    *hp = hv;
  } else {
    float4 hv = *hp;
    hv.x += theta_k * fn.x;
    hv.y += theta_k * fn.y;
    hv.z += theta_k * fn.z;
    hv.w += theta_k * fn.w;
    *hp = hv;
  }
}

// ---------------------------------------------------------------------------
// Host: feat = d_in[0] (f32, N*64), src = d_in[1] (i32, E), dst = d_in[2]
// d_out = h (f32, N*64).  Workspace: dinv | feat_cur | agg  (~52 MB).
// All work on `stream`; only capture-safe APIs (hipMemsetAsync).
// ---------------------------------------------------------------------------
extern "C" void kernel_launch(void* const* d_in, const int* in_sizes, int n_in,
                              void* d_out, int out_size, void* d_ws,
                              size_t ws_size, hipStream_t stream) {
  (void)n_in;
  (void)out_size;
  (void)ws_size;
  const float* feat = (const float*)d_in[0];
  const int* src = (const int*)d_in[1];
  const int* dst = (const int*)d_in[2];
  float* h = (float*)d_out;

  const int n_nodes = in_sizes[0] / D_FEAT;
  const int n_edges = in_sizes[1];
  const size_t feat_bytes = (size_t)n_nodes * D_FEAT * sizeof(float);

  char* ws = (char*)d_ws;
  float* dinv = (float*)ws;  // n_nodes floats (holds deg, then d^-1/2)
  size_t off = ((size_t)n_nodes * sizeof(float) + 255) & ~(size_t)255;
  float* feat_cur = (float*)(ws + off);
  float* agg = (float*)(ws + off + feat_bytes);

  const float TH[4] = {1.0f, -0.8f, 0.4f, -0.1f};

  // degree + normalization
  hipMemsetAsync(dinv, 0, (size_t)n_nodes * sizeof(float), stream);
  pc_deg<<<(n_edges + 255) / 256, 256, 0, stream>>>(dst, dinv, n_edges);
  pc_dinv<<<(n_nodes + 255) / 256, 256, 0, stream>>>(dinv, n_nodes);

  const int upd_items = n_nodes * (D_FEAT / 4);  // 1.6M float4 chunks
  const int ntiles = (n_edges + TILE_E - 1) / TILE_E;  // 2344 tiles
  int scat_blocks = (ntiles + 1) / 2;  // ~2 tiles/block: double-buffer overlap
  if (scat_blocks < 1) scat_blocks = 1;

  const float* fin = feat;  // iteration 1 reads the (immutable) input
  for (int k = 1; k <= 3; ++k) {
    hipMemsetAsync(agg, 0, feat_bytes, stream);
    pc_scatter<<<scat_blocks, SCAT_BLOCK, 0, stream>>>(fin, dinv, src, dst,
                                                       agg, n_edges);
    pc_update<<<(upd_items + 255) / 256, 256, 0, stream>>>(
        fin, agg, dinv, feat_cur, h, TH[0], TH[k], (k == 1) ? 1 : 0,
        upd_items);
    fin = feat_cur;  // subsequent iterations update feat_cur in place
  }
}